// SCSSystem_48369921688137
// MI455X (gfx1250) — compile-verified
//
#include <hip/hip_runtime.h>

// ---------------------------------------------------------------------------
// SCSSystem fused kernel for MI455X (gfx1250, wave32)
//   out[0] = axonal[8,32,256,256]  (per-pixel 8x8 connection matmul)
//   out[1] = grid  [8,32,256,256]  (constant 8x8 multi-scale matmul -> WMMA)
// Bandwidth-bound (~200MB @ 23.3TB/s ~= 9us floor): single pass over spikes,
// both outputs fused, all global traffic non-temporal.
// ---------------------------------------------------------------------------

typedef float v2f __attribute__((ext_vector_type(2)));
typedef float v8f __attribute__((ext_vector_type(8)));

#define N_NODES 8
#define HW      65536      // 256*256
#define BHW     2097152    // 32*256*256  (columns per node)
#define NPAIRS  24
#define PIX_PER_BLOCK 32
#define CSTRIDE 33         // padded pixel stride: half-waves land on disjoint banks
#define THREADS 256

__global__ __launch_bounds__(THREADS)
void scs_fused_kernel(const float* __restrict__ spikes,      // [8][BHW]
                      const float* __restrict__ E,           // [8][HW]
                      const float* __restrict__ conn_weight, // [24][HW]
                      const int*   __restrict__ conn_src,    // [24]
                      const int*   __restrict__ conn_dst,    // [24]
                      const float* __restrict__ scale_w,     // [3]
                      float* __restrict__ out)               // [2][8][BHW]
{
    __shared__ float Cmat[64 * CSTRIDE];   // [d*8+s][pix] fused per-pixel matrix

    const int tid  = threadIdx.x;
    const int lane = tid & 31;
    const int wave = tid >> 5;
    const int n    = lane & 15;            // column within 16-wide tile
    const int hi   = lane >> 4;            // half-wave id (0/1)
    const int hw0  = blockIdx.x * PIX_PER_BLOCK;

    // ---------------- phase 1: build C[d][s][pix] in LDS -------------------
    for (int i = tid; i < 64 * CSTRIDE; i += THREADS) Cmat[i] = 0.0f;
    __syncthreads();

    if (tid < PIX_PER_BLOCK) {
        const int pix = tid;
        #pragma unroll
        for (int p = 0; p < NPAIRS; ++p) {
            float cw = __builtin_nontemporal_load(conn_weight + p * HW + hw0 + pix);
            int row  = conn_dst[p] * 8 + conn_src[p];
            Cmat[row * CSTRIDE + pix] += cw;          // pix-private: no race
        }
        #pragma unroll
        for (int s = 0; s < N_NODES; ++s) {
            float f = 1.5f * __builtin_nontemporal_load(E + s * HW + hw0 + pix) - 0.5f;
            #pragma unroll
            for (int d = 0; d < N_NODES; ++d)
                Cmat[(d * 8 + s) * CSTRIDE + pix] *= f;
        }
    }
    __syncthreads();

    // ------------- constant multi-scale matrix as WMMA A fragments ---------
    // A is 16x8 (rows 8..15 zero-padded), consumed as two 16x4 K-chunks.
    // Fragment layout (ISA 7.12.2): lane L holds M=L%16, K = v + 2*(L/16).
    const float w0 = scale_w[0], w1 = scale_w[1], w2 = scale_w[2];
    v2f afrag[2];
    #pragma unroll
    for (int c = 0; c < 2; ++c) {
        #pragma unroll
        for (int v = 0; v < 2; ++v) {
            int s    = 4 * c + 2 * hi + v;
            int diff = (n > s) ? (n - s) : (s - n);
            float a  = (diff == 2) ? w0 : (diff == 3) ? w1 : (diff == 5) ? w2 : 0.0f;
            if (n >= N_NODES) a = 0.0f;               // pad rows 8..15
            afrag[c][v] = a;
        }
    }

    const size_t grid_off = (size_t)N_NODES * BHW;    // second output slab

    // ---------------- phase 2: 8 tiles of 16 columns per wave --------------
    for (int i = 0; i < 8; ++i) {
        const int t    = wave + 8 * i;                // 0..63
        const int b    = t >> 1;
        const int px0  = (t & 1) * 16;
        const size_t x0 = (size_t)b * HW + (size_t)(hw0 + px0);

        // B fragments: lane holds spikes rows s = 4c + 2*hi + v at column x0+n.
        // Each spike element is loaded exactly once across the grid -> NT.
        v2f bfrag[2];
        #pragma unroll
        for (int c = 0; c < 2; ++c)
            #pragma unroll
            for (int v = 0; v < 2; ++v) {
                int s = 4 * c + 2 * hi + v;
                bfrag[c][v] = __builtin_nontemporal_load(
                    spikes + (size_t)s * BHW + x0 + (size_t)n);
            }

        // Prefetch next tile's spike rows while this tile computes.
        if (i < 7) {
            int t2 = wave + 8 * (i + 1);
            size_t x2 = (size_t)(t2 >> 1) * HW + (size_t)(hw0 + (t2 & 1) * 16);
            #pragma unroll
            for (int c = 0; c < 2; ++c) {
                int s = 4 * c + 2 * hi;
                __builtin_prefetch(spikes + (size_t)s * BHW + x2 + (size_t)n, 0, 0);
            }
        }

        // Exchange complementary spike rows with partner half-wave (XOR 16):
        // ds_swizzle SWAPX16 (group-of-32: xor=0x10, and=0x1f) -> imm 0x401F.
        float oth[2][2];
        #pragma unroll
        for (int c = 0; c < 2; ++c)
            #pragma unroll
            for (int v = 0; v < 2; ++v) {
                int iv = __builtin_amdgcn_ds_swizzle(__float_as_int(bfrag[c][v]), 0x401F);
                oth[c][v] = __int_as_float(iv);
            }

        // grid = A x spikes : two K=4 WMMA steps (full-EXEC point, no branch).
        v8f acc = {};
        acc = __builtin_amdgcn_wmma_f32_16x16x4_f32(
            false, afrag[0], false, bfrag[0], (short)0, acc, false, false);
        acc = __builtin_amdgcn_wmma_f32_16x16x4_f32(
            false, afrag[1], false, bfrag[1], (short)0, acc, false, false);

        // D layout: VGPR r, lanes 0-15 -> row r (rows 8..15 are the zero pad).
        if (hi == 0) {
            #pragma unroll
            for (int r = 0; r < 8; ++r)
                __builtin_nontemporal_store(
                    acc[r], out + grid_off + (size_t)r * BHW + x0 + (size_t)n);
        }

        // Assemble all 8 spike rows for this lane's column from own+partner.
        float spk[8];
        #pragma unroll
        for (int s = 0; s < 8; ++s) {
            int c = s >> 2, v = s & 1;
            spk[s] = (((s & 3) >> 1) == hi) ? bfrag[c][v] : oth[c][v];
        }

        // axonal: each half-wave produces 4 of the 8 destination nodes.
        const int pix = px0 + n;
        #pragma unroll
        for (int dd = 0; dd < 4; ++dd) {
            const int d = 4 * hi + dd;
            float a2 = 0.0f;
            #pragma unroll
            for (int s = 0; s < 8; ++s)
                a2 = fmaf(Cmat[(d * 8 + s) * CSTRIDE + pix], spk[s], a2);
            __builtin_nontemporal_store(a2, out + (size_t)d * BHW + x0 + (size_t)n);
        }
    }
}

extern "C" void kernel_launch(void* const* d_in, const int* in_sizes, int n_in,
                              void* d_out, int out_size, void* d_ws, size_t ws_size,
                              hipStream_t stream) {
    (void)in_sizes; (void)n_in; (void)out_size; (void)d_ws; (void)ws_size;
    const float* spikes = (const float*)d_in[0];
    const float* E      = (const float*)d_in[1];
    const float* cw     = (const float*)d_in[2];
    const int*   csrc   = (const int*)d_in[3];
    const int*   cdst   = (const int*)d_in[4];
    const float* sw     = (const float*)d_in[5];

    dim3 grid(HW / PIX_PER_BLOCK);   // 2048 blocks
    dim3 block(THREADS);             // 8 waves each
    hipLaunchKernelGGL(scs_fused_kernel, grid, block, 0, stream,
                       spikes, E, cw, csrc, cdst, sw, (float*)d_out);
}